// GCN_23390391894415
// MI455X (gfx1250) — compile-verified
//
#include <hip/hip_runtime.h>
#include <hip/hip_bf16.h>

#define MD   64
#define LNEPS 1e-3f

typedef __attribute__((ext_vector_type(2))) float v2f;
typedef __attribute__((ext_vector_type(8))) float v8f;

__device__ __forceinline__ void atomic_add_f32(float* p, float v) {
    unsafeAtomicAdd(p, v);   // -> global_atomic_add_f32 on gfx1250
}

// ---------------------------------------------------------------- utilities
__global__ void fill0_kernel(float* __restrict__ p, int n) {
    int i = blockIdx.x * blockDim.x + threadIdx.x;
    if (i < n) p[i] = 0.0f;
}

__global__ void degree_kernel(const int* __restrict__ src_ab, const int* __restrict__ dst_ab,
                              const int* __restrict__ src_ba, const int* __restrict__ dst_ba,
                              float* __restrict__ cnt, int nE, int N) {
    int i = blockIdx.x * blockDim.x + threadIdx.x;
    if (i < nE) {
        atomic_add_f32(cnt + src_ab[i],          1.0f);  // deg_out A (rel A->B)
        atomic_add_f32(cnt + N + dst_ab[i],      1.0f);  // deg_in  B
        atomic_add_f32(cnt + 2 * N + src_ba[i],  1.0f);  // deg_out B (rel B->A)
        atomic_add_f32(cnt + 3 * N + dst_ba[i],  1.0f);  // deg_in  A
    }
}

__global__ void rsqrt_deg_kernel(float* __restrict__ p, int n) {
    int i = blockIdx.x * blockDim.x + threadIdx.x;
    if (i < n) p[i] = rsqrtf(fmaxf(p[i], 1.0f));
}

__global__ void init_h_kernel(const float* __restrict__ embA, const float* __restrict__ embB,
                              float* __restrict__ hA, float* __restrict__ hB, int nd) {
    int i = blockIdx.x * blockDim.x + threadIdx.x;
    if (i < nd) {
        int d = i & (MD - 1);
        hA[i] = embA[d];
        hB[i] = embB[d];
    }
}

// ------------------------------------------------------- edge gather/scatter
// agg[dst] += h[src] * rs_out[src].  One wave = 2 edges, 16 lanes x float4 each.
__global__ __launch_bounds__(256)
void scatter_kernel(const float* __restrict__ h, const int* __restrict__ src,
                    const int* __restrict__ dst, const float* __restrict__ rs_out,
                    float* __restrict__ agg, int nE) {
    const int lane   = threadIdx.x & 31;
    const int wave   = (int)((blockIdx.x * blockDim.x + threadIdx.x) >> 5);
    const int nWaves = (int)((gridDim.x * blockDim.x) >> 5);
    const int sub    = lane >> 4;             // 0/1: which edge of the pair
    const int c      = (lane & 15) * 4;       // column base (4 floats)

    for (long long e0 = (long long)wave * 2; e0 < nE; e0 += (long long)nWaves * 2) {
        long long e = e0 + sub;
        if (e < nE) {
            int s = src[e];
            int d = dst[e];
            float sc = rs_out[s];
            float4 v = *reinterpret_cast<const float4*>(h + (size_t)s * MD + c);
            float* p = agg + (size_t)d * MD + c;
            atomic_add_f32(p + 0, v.x * sc);
            atomic_add_f32(p + 1, v.y * sc);
            atomic_add_f32(p + 2, v.z * sc);
            atomic_add_f32(p + 3, v.w * sc);
        }
    }
}

// ----------------------------------------------- fused GEMM + ReLU + residual + LN
// h = LayerNorm(h + relu((agg * rs_in[:,None]) @ W + b), beta)      (in place)
// Per wave: one 16-row strip.  WMMA f32 16x16x4, K loop of 16, 4 N-tiles.
__global__ __launch_bounds__(256)
void gemm_ln_kernel(const float* __restrict__ agg, const float* __restrict__ rs_in,
                    const float* __restrict__ W,    // [64,64] row-major (k,n)
                    const float* __restrict__ bias, const float* __restrict__ beta,
                    float* __restrict__ h, int nRows) {
    __shared__ float2 sWp[32 * 64];      // (k/2, n): {W[2k][n], W[2k+1][n]}  (16 KB)
    __shared__ float  sBias[MD];
    __shared__ float  sBeta[MD];
    __shared__ float  sOut[8][16 * MD];  // per-wave 16x64 tile (32 KB)

    const int tid  = threadIdx.x;
    const int lane = tid & 31;
    const int wave = tid >> 5;

    // cooperative load of W in k-paired layout -> B fragment = one ds_load_b64
    for (int i = tid; i < 32 * 64; i += 256) {
        int kk = i >> 6;      // k-pair 0..31
        int n  = i & 63;
        float2 w2;
        w2.x = W[(2 * kk) * MD + n];
        w2.y = W[(2 * kk + 1) * MD + n];
        sWp[i] = w2;
    }
    if (tid < MD) { sBias[tid] = bias[tid]; sBeta[tid] = beta[tid]; }
    __syncthreads();

    const int nStrips = nRows >> 4;
    const int strip   = blockIdx.x * 8 + wave;
    if (strip >= nStrips) return;

    const int   base   = strip * 16;
    const int   nlo    = lane & 15;
    const int   khalf  = (lane >> 4) << 1;          // 0 or 2
    const int   arow   = base + nlo;                // A: lanes 0-15 M, 16-31 same M hi-K
    const float rscale = rs_in[arow];
    const float* Arow  = agg + (size_t)arow * MD;

    v8f acc0 = {}, acc1 = {}, acc2 = {}, acc3 = {};

#pragma unroll
    for (int k0 = 0; k0 < MD; k0 += 4) {
        float2 a2 = *reinterpret_cast<const float2*>(Arow + k0 + khalf);
        v2f av; av.x = a2.x * rscale; av.y = a2.y * rscale;
        const int kp = ((k0 + khalf) >> 1) * 64 + nlo;   // index into sWp
        float2 b0 = sWp[kp + 0];
        float2 b1 = sWp[kp + 16];
        float2 b2 = sWp[kp + 32];
        float2 b3 = sWp[kp + 48];
        v2f bv0; bv0.x = b0.x; bv0.y = b0.y;
        v2f bv1; bv1.x = b1.x; bv1.y = b1.y;
        v2f bv2; bv2.x = b2.x; bv2.y = b2.y;
        v2f bv3; bv3.x = b3.x; bv3.y = b3.y;
        acc0 = __builtin_amdgcn_wmma_f32_16x16x4_f32(false, av, false, bv0, (short)0, acc0, false, false);
        acc1 = __builtin_amdgcn_wmma_f32_16x16x4_f32(false, av, false, bv1, (short)0, acc1, false, false);
        acc2 = __builtin_amdgcn_wmma_f32_16x16x4_f32(false, av, false, bv2, (short)0, acc2, false, false);
        acc3 = __builtin_amdgcn_wmma_f32_16x16x4_f32(false, av, false, bv3, (short)0, acc3, false, false);
    }

    // spill C tiles to LDS with bias + relu
    // C layout: VGPR r -> M = r (lanes 0-15) / r+8 (lanes 16-31), N = t*16 + (lane&15)
    float* myOut = &sOut[wave][0];
    const int mbase = (lane < 16) ? 0 : 8;
#pragma unroll
    for (int r = 0; r < 8; ++r) {
        int mm = mbase + r;
        myOut[mm * MD + 0  + nlo] = fmaxf(acc0[r] + sBias[0  + nlo], 0.0f);
        myOut[mm * MD + 16 + nlo] = fmaxf(acc1[r] + sBias[16 + nlo], 0.0f);
        myOut[mm * MD + 32 + nlo] = fmaxf(acc2[r] + sBias[32 + nlo], 0.0f);
        myOut[mm * MD + 48 + nlo] = fmaxf(acc3[r] + sBias[48 + nlo], 0.0f);
    }
    asm volatile("s_wait_dscnt 0" ::: "memory");   // wave-private tile, LDS in-order/wave

    // residual + LayerNorm: 2 lanes per row (32 elements each)
    const int   r   = lane >> 1;
    const int   c0  = (lane & 1) * 32;
    float* hrow = h + (size_t)(base + r) * MD;
    float4 xv[8];
    float sum = 0.0f, sumsq = 0.0f;
#pragma unroll
    for (int j = 0; j < 8; ++j) {
        float4 o  = *reinterpret_cast<const float4*>(&myOut[r * MD + c0 + 4 * j]);
        float4 hv = *reinterpret_cast<const float4*>(hrow + c0 + 4 * j);
        float4 x;
        x.x = o.x + hv.x; x.y = o.y + hv.y; x.z = o.z + hv.z; x.w = o.w + hv.w;
        xv[j] = x;
        sum   += x.x + x.y + x.z + x.w;
        sumsq += x.x * x.x + x.y * x.y + x.z * x.z + x.w * x.w;
    }
    sum   += __shfl_xor(sum,   1, 32);
    sumsq += __shfl_xor(sumsq, 1, 32);
    const float mu  = sum * (1.0f / MD);
    const float var = sumsq * (1.0f / MD) - mu * mu;
    const float inv = rsqrtf(var + LNEPS);
#pragma unroll
    for (int j = 0; j < 8; ++j) {
        float4 x = xv[j];
        float4 o;
        o.x = (x.x - mu) * inv + sBeta[c0 + 4 * j + 0];
        o.y = (x.y - mu) * inv + sBeta[c0 + 4 * j + 1];
        o.z = (x.z - mu) * inv + sBeta[c0 + 4 * j + 2];
        o.w = (x.w - mu) * inv + sBeta[c0 + 4 * j + 3];
        *reinterpret_cast<float4*>(hrow + c0 + 4 * j) = o;
    }
}

// ---------------------------------------------------------------- launcher
extern "C" void kernel_launch(void* const* d_in, const int* in_sizes, int n_in,
                              void* d_out, int out_size, void* d_ws, size_t ws_size,
                              hipStream_t stream) {
    const int*   src_ab = (const int*)d_in[0];
    const int*   dst_ab = (const int*)d_in[1];
    const int*   src_ba = (const int*)d_in[2];
    const int*   dst_ba = (const int*)d_in[3];
    const float* emb_A  = (const float*)d_in[4];
    const float* emb_B  = (const float*)d_in[5];
    const float* W_ab   = (const float*)d_in[6];
    const float* b_ab   = (const float*)d_in[7];
    const float* W_ba   = (const float*)d_in[8];
    const float* b_ba   = (const float*)d_in[9];
    const float* beta_A = (const float*)d_in[10];
    const float* beta_B = (const float*)d_in[11];

    const int E = in_sizes[0];
    const int N = out_size / (2 * MD);
    const int L = in_sizes[6] / (MD * MD);

    const size_t ND = (size_t)N * MD;
    float* ws       = (float*)d_ws;
    float* aggA     = ws;                 // [N, 64]
    float* aggB     = ws + ND;            // [N, 64]  (contiguous with aggA)
    float* cnt      = ws + 2 * ND;        // [4, N] degree counts -> rsqrt scales
    float* rs_out_A = cnt;
    float* rs_in_B  = cnt + N;
    float* rs_out_B = cnt + 2 * (size_t)N;
    float* rs_in_A  = cnt + 3 * (size_t)N;

    float* hA = (float*)d_out;            // output layout: [hA ; hB]
    float* hB = hA + ND;

    const int nStrips    = N / 16;
    const int gemmBlocks = (nStrips + 7) / 8;

    fill0_kernel<<<(4 * N + 255) / 256, 256, 0, stream>>>(cnt, 4 * N);
    degree_kernel<<<(E + 255) / 256, 256, 0, stream>>>(src_ab, dst_ab, src_ba, dst_ba, cnt, E, N);
    rsqrt_deg_kernel<<<(4 * N + 255) / 256, 256, 0, stream>>>(cnt, 4 * N);
    init_h_kernel<<<((int)ND + 255) / 256, 256, 0, stream>>>(emb_A, emb_B, hA, hB, (int)ND);

    for (int l = 0; l < L; ++l) {
        fill0_kernel<<<((int)(2 * ND) + 255) / 256, 256, 0, stream>>>(aggA, (int)(2 * ND));
        // A -> B : aggB += norm(hA)[src_ab] scattered to dst_ab
        scatter_kernel<<<2048, 256, 0, stream>>>(hA, src_ab, dst_ab, rs_out_A, aggB, E);
        // B -> A : aggA += norm(hB)[src_ba] scattered to dst_ba
        scatter_kernel<<<2048, 256, 0, stream>>>(hB, src_ba, dst_ba, rs_out_B, aggA, E);
        // hA = LN(hA + relu((aggA * rs_in_A) @ W_ba + b_ba), beta_A)
        gemm_ln_kernel<<<gemmBlocks, 256, 0, stream>>>(aggA, rs_in_A, W_ba + (size_t)l * MD * MD,
                                                       b_ba + (size_t)l * MD, beta_A + (size_t)l * MD, hA, N);
        // hB = LN(hB + relu((aggB * rs_in_B) @ W_ab + b_ab), beta_B)
        gemm_ln_kernel<<<gemmBlocks, 256, 0, stream>>>(aggB, rs_in_B, W_ab + (size_t)l * MD * MD,
                                                       b_ab + (size_t)l * MD, beta_B + (size_t)l * MD, hB, N);
    }
}